// GraphActor_1984274891290
// MI455X (gfx1250) — compile-verified
//
#include <hip/hip_runtime.h>
#include <hip/hip_bf16.h>
#include <math.h>

#define N_NODES 40000
#define N_EDGES 640000
#define HID     128
#define N_SGEN  4096

typedef float v2f __attribute__((ext_vector_type(2)));
typedef float v8f __attribute__((ext_vector_type(8)));

// ---------------------------------------------------------------------------
// Generic zero-fill
// ---------------------------------------------------------------------------
__global__ void k_zero(float* p, int n) {
    int t = blockIdx.x * blockDim.x + threadIdx.x;
    if (t < n) p[t] = 0.0f;
}

// ---------------------------------------------------------------------------
// Degree accumulation: deg[dst] += 1 per edge (into dinv buffer, pre-zeroed)
// ---------------------------------------------------------------------------
__global__ void k_degree(const int* __restrict__ dst, float* __restrict__ deg) {
    int e = blockIdx.x * blockDim.x + threadIdx.x;
    if (e < N_EDGES) unsafeAtomicAdd(&deg[dst[e]], 1.0f);
}

// deg -> (deg + 1)^-1/2   (self-loop included; deg+1 >= 1 so always valid)
__global__ void k_dinv(float* __restrict__ d) {
    int t = blockIdx.x * blockDim.x + threadIdx.x;
    if (t < N_NODES) d[t] = 1.0f / sqrtf(d[t] + 1.0f);
}

// ---------------------------------------------------------------------------
// WMMA fp32 GEMM:  C[M x 128] = A[M x 128] @ B[128 x 128] (+ bias)
// One wave computes one 16x16 tile of C, accumulating K in steps of 4 with
// V_WMMA_F32_16X16X4_F32. M must be a multiple of 16 (40000 = 2500*16).
// grid = (M/16, 2), block = 128 (4 waves); wave w of block-row y covers
// N-tile (y*4 + w).
//
// Fragment layouts (wave32, per ISA 7.12.2):
//   A 16x4:  lanes 0-15 -> M=lane, VGPR0/1 = K0/K1; lanes 16-31 -> K2/K3
//   B 4x16:  lanes 0-15 -> N=lane, VGPR0/1 = K0/K1; lanes 16-31 -> K2/K3
//   C 16x16: VGPR r, lanes 0-15 -> M=r, N=lane; lanes 16-31 -> M=r+8
// ---------------------------------------------------------------------------
__global__ void k_gemm_wmma(const float* __restrict__ A,
                            const float* __restrict__ B,
                            const float* __restrict__ bias,
                            float* __restrict__ C,
                            int hasBias) {
    const int lane   = threadIdx.x & 31;
    const int wave   = threadIdx.x >> 5;
    const int tileM  = blockIdx.x * 16;
    const int tileN  = (blockIdx.y * 4 + wave) * 16;
    const int l15    = lane & 15;
    const int koff   = (lane >> 4) * 2;          // 0 or 2

    const float* arow = A + (size_t)(tileM + l15) * HID + koff;
    const float* bcol = B + (size_t)koff * HID + (tileN + l15);

    v8f acc = {};
#pragma unroll 4
    for (int k = 0; k < HID; k += 4) {
        v2f a = *(const v2f*)(arow + k);         // A[m][k+koff .. +1], 8B aligned
        v2f b;
        b.x = bcol[(size_t)k * HID];             // B[k+koff  ][n]
        b.y = bcol[(size_t)(k + 1) * HID];       // B[k+koff+1][n]
        acc = __builtin_amdgcn_wmma_f32_16x16x4_f32(
                  false, a, false, b, (short)0, acc, false, false);
    }

    const float bv = hasBias ? bias[tileN + l15] : 0.0f;
    const int   m0 = tileM + (lane >> 4) * 8;
    float* crow = C + (size_t)m0 * HID + tileN + l15;
#pragma unroll
    for (int r = 0; r < 8; ++r)
        crow[(size_t)r * HID] = acc[r] + bv;
}

// ---------------------------------------------------------------------------
// Edge scatter: agg[dst] += xw[src] * dinv[src]*dinv[dst]
// One wave per edge, each lane handles 4 contiguous features (32*4 = 128).
// ---------------------------------------------------------------------------
__global__ void k_edge_scatter(const float* __restrict__ xw,
                               const int* __restrict__ src_a,
                               const int* __restrict__ dst_a,
                               const float* __restrict__ dinv,
                               float* __restrict__ agg) {
    int t = blockIdx.x * blockDim.x + threadIdx.x;
    int e = t >> 5;
    if (e >= N_EDGES) return;
    int c = t & 31;
    int s = src_a[e];
    int d = dst_a[e];
    float norm = dinv[s] * dinv[d];
    const float4 v = ((const float4*)(xw + (size_t)s * HID))[c];
    float* out = agg + (size_t)d * HID + c * 4;
    unsafeAtomicAdd(out + 0, v.x * norm);
    unsafeAtomicAdd(out + 1, v.y * norm);
    unsafeAtomicAdd(out + 2, v.z * norm);
    unsafeAtomicAdd(out + 3, v.w * norm);
}

// ---------------------------------------------------------------------------
// Conv finalize:  h = relu(agg + xw * dinv[node]^2 + bias[f])   (in place on agg)
// ---------------------------------------------------------------------------
__global__ void k_conv_finalize(float* __restrict__ agg,
                                const float* __restrict__ xw,
                                const float* __restrict__ dinv,
                                const float* __restrict__ bias) {
    int t = blockIdx.x * blockDim.x + threadIdx.x;
    if (t >= N_NODES * HID) return;
    int node = t >> 7;
    int f    = t & 127;
    float di = dinv[node];
    float v  = agg[t] + xw[t] * di * di + bias[f];
    agg[t]   = fmaxf(v, 0.0f);
}

// ---------------------------------------------------------------------------
// Head: one wave per sgen row. mu/ls dot products over 128 features,
// rsample + tanh, per-row logp contribution to workspace.
// ---------------------------------------------------------------------------
__global__ void k_head(const float* __restrict__ h,
                       const int* __restrict__ sgen_map,
                       const float* __restrict__ noise,
                       const float* __restrict__ mu_w, const float* __restrict__ mu_b,
                       const float* __restrict__ ls_w, const float* __restrict__ ls_b,
                       float* __restrict__ a_out,
                       float* __restrict__ logp_part) {
    int gw   = (blockIdx.x * blockDim.x + threadIdx.x) >> 5;
    int lane = threadIdx.x & 31;
    if (gw >= N_SGEN) return;

    const float* z = h + (size_t)sgen_map[gw] * HID;
    float macc = 0.0f, lacc = 0.0f;
#pragma unroll
    for (int i = lane; i < HID; i += 32) {
        float zv = z[i];
        macc += zv * mu_w[i];
        lacc += zv * ls_w[i];
    }
#pragma unroll
    for (int off = 16; off > 0; off >>= 1) {
        macc += __shfl_xor(macc, off, 32);
        lacc += __shfl_xor(lacc, off, 32);
    }
    if (lane == 0) {
        float mu  = macc + mu_b[0];
        float ls  = fminf(fmaxf(lacc + ls_b[0], -20.0f), 2.0f);
        float sd  = expf(ls);
        float u   = mu + sd * noise[gw];
        float a   = tanhf(u);
        a_out[gw] = a;
        float dm  = u - mu;
        float lp  = -(dm * dm) / (2.0f * sd * sd) - ls
                    - 0.91893853320467274178f            /* 0.5*log(2*pi) */
                    - logf(1.0f - a * a + 1e-6f);        /* minus log-jacobian */
        logp_part[gw] = lp;
    }
}

// ---------------------------------------------------------------------------
// Deterministic fixed-order reduction of the 4096 logp terms (single block).
// ---------------------------------------------------------------------------
__global__ void k_reduce_logp(const float* __restrict__ part, float* __restrict__ out) {
    __shared__ float sm[256];
    int t = threadIdx.x;
    float s = 0.0f;
#pragma unroll
    for (int i = t; i < N_SGEN; i += 256) s += part[i];
    sm[t] = s;
    __syncthreads();
    for (int off = 128; off > 0; off >>= 1) {
        if (t < off) sm[t] += sm[t + off];
        __syncthreads();
    }
    if (t == 0) out[0] = sm[0];
}

// ---------------------------------------------------------------------------
extern "C" void kernel_launch(void* const* d_in, const int* in_sizes, int n_in,
                              void* d_out, int out_size, void* d_ws, size_t ws_size,
                              hipStream_t stream) {
    const float* x         = (const float*)d_in[0];
    /* d_in[1] edge_attr, d_in[7] edge_in_w, d_in[8] edge_in_b: dead in reference */
    const float* noise     = (const float*)d_in[2];
    const int*   edge_idx  = (const int*)  d_in[3];
    const int*   sgen_map  = (const int*)  d_in[4];
    const float* node_in_w = (const float*)d_in[5];
    const float* node_in_b = (const float*)d_in[6];
    const float* conv1_w   = (const float*)d_in[9];
    const float* conv1_b   = (const float*)d_in[10];
    const float* conv2_w   = (const float*)d_in[11];
    const float* conv2_b   = (const float*)d_in[12];
    const float* mu_w      = (const float*)d_in[13];
    const float* mu_b      = (const float*)d_in[14];
    const float* ls_w      = (const float*)d_in[15];
    const float* ls_b      = (const float*)d_in[16];

    const int* e_src = edge_idx;            // edge_index[0, :]
    const int* e_dst = edge_idx + N_EDGES;  // edge_index[1, :]

    float* ws   = (float*)d_ws;
    float* bufA = ws;                               // 5,120,000 floats
    float* bufB = ws + (size_t)N_NODES * HID;       // 5,120,000 floats
    float* dinv = bufB + (size_t)N_NODES * HID;     // 40,000 floats
    float* lpp  = dinv + N_NODES;                   // 4,096 floats

    float* out  = (float*)d_out;                    // [0..4095]=a_tanh, [4096]=logp

    const int NF = N_NODES * HID;                   // 5,120,000

    // 1) degree -> dinv
    k_zero<<<(N_NODES + 255) / 256, 256, 0, stream>>>(dinv, N_NODES);
    k_degree<<<(N_EDGES + 255) / 256, 256, 0, stream>>>(e_dst, dinv);
    k_dinv<<<(N_NODES + 255) / 256, 256, 0, stream>>>(dinv);

    // 2) h = x @ node_in_w + b   -> bufA
    dim3 ggrid(N_NODES / 16, 2);
    k_gemm_wmma<<<ggrid, 128, 0, stream>>>(x, node_in_w, node_in_b, bufA, 1);

    // 3) conv1
    k_gemm_wmma<<<ggrid, 128, 0, stream>>>(bufA, conv1_w, nullptr, bufB, 0);
    k_zero<<<(NF + 255) / 256, 256, 0, stream>>>(bufA, NF);
    k_edge_scatter<<<(N_EDGES * 32) / 256, 256, 0, stream>>>(bufB, e_src, e_dst, dinv, bufA);
    k_conv_finalize<<<(NF + 255) / 256, 256, 0, stream>>>(bufA, bufB, dinv, conv1_b);

    // 4) conv2
    k_gemm_wmma<<<ggrid, 128, 0, stream>>>(bufA, conv2_w, nullptr, bufB, 0);
    k_zero<<<(NF + 255) / 256, 256, 0, stream>>>(bufA, NF);
    k_edge_scatter<<<(N_EDGES * 32) / 256, 256, 0, stream>>>(bufB, e_src, e_dst, dinv, bufA);
    k_conv_finalize<<<(NF + 255) / 256, 256, 0, stream>>>(bufA, bufB, dinv, conv2_b);

    // 5) head + deterministic logp reduction
    k_head<<<(N_SGEN * 32) / 256, 256, 0, stream>>>(bufA, sgen_map, noise,
                                                    mu_w, mu_b, ls_w, ls_b,
                                                    out, lpp);
    k_reduce_logp<<<1, 256, 0, stream>>>(lpp, out + N_SGEN);
}